// PFPC_56092272886307
// MI455X (gfx1250) — compile-verified
//
#include <hip/hip_runtime.h>
#include <hip/hip_bf16.h>
#include <math.h>

// ---------------------------------------------------------------------------
// InfoNCE Gaussian-decoder MI bounds, fused for MI455X (gfx1250, wave32).
//   scores = [y^2, y] @ [(-.5/var) ; (mean/var)]^T + c_j   (M=N=8192, K=256)
// fp32 WMMA (V_WMMA_F32_16X16X4_F32): exact fp32 since logits feed logsumexp.
// All B-side operands (and LDS A tiles) are stored in WMMA-fragment K-order
//   perm(k) = ((k>>1)&1)*128 + (k>>2)*2 + (k&1)
// so one b128 load feeds two WMMAs (k-halves are contiguous per lane).
// Score matrix is never materialized: per-lane online logsumexp in registers,
// butterfly-merged across lanes once after the full column sweep.
// ---------------------------------------------------------------------------

typedef float v2f __attribute__((ext_vector_type(2)));
typedef float v4f __attribute__((ext_vector_type(4)));
typedef float v8f __attribute__((ext_vector_type(8)));

#define NROWS 8192
#define XD    256
#define YD    128
#define THRESH 1e-6f

__device__ __forceinline__ v8f wmma_f32x4(v2f a, v2f b, v8f c) {
  return __builtin_amdgcn_wmma_f32_16x16x4_f32(
      false, a, false, b, (short)0, c, false, false);
}

__device__ __forceinline__ int permk(int k) {       // fragment K-order
  return (((k >> 1) & 1) << 7) + (((k >> 2)) << 1) + (k & 1);
}

// ---------------------------------------------------------------------------
// Weight pack: Wp[n][perm(k)] = W[k][n]   (K = 256 for every layer)
// ---------------------------------------------------------------------------
__global__ __launch_bounds__(256) void pack_w(const float* __restrict__ W,
                                              float* __restrict__ Wp, int N) {
  const int idx = blockIdx.x * 256 + threadIdx.x;   // over 256*N
  const int k = idx / N, n = idx % N;
  Wp[(size_t)n * 256 + permk(k)] = W[idx];
}

// ---------------------------------------------------------------------------
// One 16-row x N-col fp32 layer. sIn: LDS 16x256 in perm-K order.
// Wp: packed weights (N x 256 perm-K). permOut: store output in perm-K order
// (when it feeds the next layer's K=256).
// ---------------------------------------------------------------------------
__device__ __forceinline__ void layer16(const float* sIn, const float* __restrict__ Wp,
                                        const float* __restrict__ bias, float* sOut,
                                        int N, bool dorelu, bool permOut,
                                        int wave, int lane) {
  const int m_a   = lane & 15;
  const int khalf = lane >> 4;
  const int n     = lane & 15;
  const float* aptr = sIn + m_a * 256 + (khalf << 7);
  const int NT = N >> 4;
  for (int ct = wave; ct < NT; ct += 8) {
    const int n0 = ct << 4;
    const float* bptr = Wp + (size_t)(n0 + n) * 256 + (khalf << 7);
    v8f acc = {};
#pragma unroll 8
    for (int p = 0; p < 128; p += 4) {              // 1 b128 -> 2 WMMAs
      const v4f a = *(const v4f*)(aptr + p);
      const v4f b = *(const v4f*)(bptr + p);
      v2f a0 = {a.x, a.y}, b0 = {b.x, b.y};
      acc = wmma_f32x4(a0, b0, acc);
      v2f a1 = {a.z, a.w}, b1 = {b.z, b.w};
      acc = wmma_f32x4(a1, b1, acc);
    }
    const float bv = bias[n0 + n];
    // perm position of output column (becomes next layer's k)
    const int pcol = (((n >> 1) & 1) << 7) + (n0 >> 1) + ((n >> 2) << 1) + (n & 1);
#pragma unroll
    for (int v = 0; v < 8; ++v) {
      float val = acc[v] + bv;
      if (dorelu) val = fmaxf(val, 0.0f);
      const int m = v + (khalf << 3);
      if (permOut) sOut[m * 256 + pcol] = val;
      else         sOut[m * N + n0 + n] = val;
    }
  }
}

// ---------------------------------------------------------------------------
// Phase 1: both MLPs + per-row decoder precompute (perm-K packed):
//   Bc[j][perm(d)]      = -0.5 / var_j
//   Bc[j][perm(128+d)]  =  mean_j / var_j      (== perm(d)+64)
//   cvec[j]             = -0.5 * (sum log var + sum mean^2 / var)
// ---------------------------------------------------------------------------
__global__ __launch_bounds__(256) void mlp_kernel(
    const float* __restrict__ x,
    const float* __restrict__ Wm1p, const float* __restrict__ bm1,
    const float* __restrict__ Wm2p, const float* __restrict__ bm2,
    const float* __restrict__ Wm3p, const float* __restrict__ bm3,
    const float* __restrict__ Wv1p, const float* __restrict__ bv1,
    const float* __restrict__ Wv2p, const float* __restrict__ bv2,
    const float* __restrict__ Wv3p, const float* __restrict__ bv3,
    float* __restrict__ Bc, float* __restrict__ cvec) {
  __shared__ float xbuf[16 * 256];
  __shared__ float buf1[16 * 256];
  __shared__ float buf2[16 * 256];
  __shared__ float meanb[16 * 128];
  __shared__ float varb[16 * 128];
  __shared__ float cred[256];

  const int tid = threadIdx.x;
  const int wave = tid >> 5, lane = tid & 31;
  const int rowbase = blockIdx.x * 16;

  for (int idx = tid; idx < 16 * 256; idx += 256) {
    const int r = idx >> 8, k = idx & 255;
    xbuf[r * 256 + permk(k)] = x[(size_t)(rowbase + r) * XD + k];
  }
  __syncthreads();

  for (int pass = 0; pass < 2; ++pass) {
    const float* W1 = pass ? Wv1p : Wm1p; const float* b1 = pass ? bv1 : bm1;
    const float* W2 = pass ? Wv2p : Wm2p; const float* b2 = pass ? bv2 : bm2;
    const float* W3 = pass ? Wv3p : Wm3p; const float* b3 = pass ? bv3 : bm3;
    layer16(xbuf, W1, b1, buf1, 256, true, true,  wave, lane);
    __syncthreads();
    layer16(buf1, W2, b2, buf2, 256, true, true,  wave, lane);
    __syncthreads();
    layer16(buf2, W3, b3, pass ? varb : meanb, 128, false, false, wave, lane);
    __syncthreads();
  }

  {
    const int r = tid >> 4, t = tid & 15;
    float csum = 0.0f;
    float* bcr = Bc + (size_t)(rowbase + r) * 256;
    for (int d = t; d < 128; d += 16) {
      float var = expf(varb[r * 128 + d]);
      if (var < THRESH) var += THRESH;              // matches reference where()
      const float inv  = 1.0f / var;
      const float mean = meanb[r * 128 + d];
      const int pq = permk(d);                      // perm(128+d) == pq + 64
      bcr[pq]      = -0.5f * inv;
      bcr[pq + 64] = mean * inv;
      csum += logf(var) + mean * mean * inv;
    }
    cred[tid] = csum;
  }
  __syncthreads();
  if (tid < 16) {
    float s = 0.0f;
    for (int t = 0; t < 16; ++t) s += cred[tid * 16 + t];
    cvec[rowbase + tid] = -0.5f * s;
  }
}

// ---------------------------------------------------------------------------
// Phase 2: fused 8192x8192x256 fp32 WMMA GEMM + online row logsumexp + diag.
// Block = 32 rows, sweeps all 8192 cols; 8 waves = 2 row-tiles x 4 col-tiles.
// Per-lane online LSE in the loop (no shuffles); one butterfly merge at end.
// ---------------------------------------------------------------------------
__global__ __launch_bounds__(256) void score_kernel(
    const float* __restrict__ y, const float* __restrict__ Bc,
    const float* __restrict__ cvec,
    float* __restrict__ rowmax, float* __restrict__ rowsum,
    float* __restrict__ diag) {
  __shared__ float Atile[32 * 256];
  __shared__ float mM[8 * 16];
  __shared__ float mS[8 * 16];

  const int tid = threadIdx.x;
  const int wave = tid >> 5, lane = tid & 31;
  const int rowtile = wave >> 2;        // 0..1
  const int coltile = wave & 3;         // 0..3
  const int m_a = lane & 15, khalf = lane >> 4, n = lane & 15;
  const int rowbase = blockIdx.x * 32;

  // Stage combined A = [y^2, y] (32 x 256) in LDS, perm-K order.
  for (int idx = tid; idx < 32 * 256; idx += 256) {
    const int r = idx >> 8, k = idx & 255;
    const float yv = y[(size_t)(rowbase + r) * YD + (k & 127)];
    Atile[r * 256 + permk(k)] = (k < 128) ? yv * yv : yv;
  }
  __syncthreads();

  const float* aptr = Atile + (rowtile * 16 + m_a) * 256 + (khalf << 7);

  float lmax[8], lsum[8];
#pragma unroll
  for (int v = 0; v < 8; ++v) { lmax[v] = -INFINITY; lsum[v] = 0.0f; }

  for (int it = 0; it < NROWS / 64; ++it) {
    const int j = it * 64 + coltile * 16 + n;       // this lane's column
    const float* bptr = Bc + (size_t)j * 256 + (khalf << 7);
    v8f acc = {};
#pragma unroll 8
    for (int p = 0; p < 128; p += 4) {              // 1 b128 -> 2 WMMAs
      const v4f a = *(const v4f*)(aptr + p);
      const v4f b = *(const v4f*)(bptr + p);
      v2f a0 = {a.x, a.y}, b0 = {b.x, b.y};
      acc = wmma_f32x4(a0, b0, acc);
      v2f a1 = {a.z, a.w}, b1 = {b.z, b.w};
      acc = wmma_f32x4(a1, b1, acc);
    }
    const float cj = cvec[j];                       // column fixed per lane

#pragma unroll
    for (int v = 0; v < 8; ++v) {
      const float val = acc[v] + cj;
      // diagonal capture (unique writer)
      const int ig = rowbase + rowtile * 16 + v + (khalf << 3);
      if (j == ig) diag[ig] = val;
      // per-lane online logsumexp (2 exps, no shuffles)
      const float nm = fmaxf(lmax[v], val);
      lsum[v] = lsum[v] * __expf(lmax[v] - nm) + __expf(val - nm);
      lmax[v] = nm;
    }
  }

  // butterfly merge across the 16 columns of this wave (within 16-lane half)
#pragma unroll
  for (int v = 0; v < 8; ++v) {
    float M = lmax[v], S = lsum[v];
    for (int msk = 1; msk < 16; msk <<= 1) {
      const float M2 = __shfl_xor(M, msk, 32);
      const float S2 = __shfl_xor(S, msk, 32);
      const float nm = fmaxf(M, M2);
      S = S * __expf(M - nm) + S2 * __expf(M2 - nm);
      M = nm;
    }
    lmax[v] = M; lsum[v] = S;
  }

  // cross-wave merge: lanes n==0 of each half hold rows (v + 8*half)
  if (n == 0) {
#pragma unroll
    for (int v = 0; v < 8; ++v) {
      const int m = v + (khalf << 3);
      mM[wave * 16 + m] = lmax[v];
      mS[wave * 16 + m] = lsum[v];
    }
  }
  __syncthreads();
  if (tid < 32) {
    const int rt = tid >> 4, row = tid & 15;
    float M = -INFINITY, S = 0.0f;
    for (int w = rt * 4; w < rt * 4 + 4; ++w) {
      const float m2 = mM[w * 16 + row], s2 = mS[w * 16 + row];
      const float nm = fmaxf(M, m2);
      S = S * __expf(M - nm) + s2 * __expf(m2 - nm);
      M = nm;
    }
    const int ig = rowbase + rt * 16 + row;
    rowmax[ig] = M;
    rowsum[ig] = S;
  }
}

// ---------------------------------------------------------------------------
// Phase 3: reduce 8192 rows -> [mi_lower, mi_upper]
// ---------------------------------------------------------------------------
__global__ __launch_bounds__(256) void reduce_kernel(
    const float* __restrict__ rowmax, const float* __restrict__ rowsum,
    const float* __restrict__ diag, float* __restrict__ out) {
  __shared__ float s1[256];
  __shared__ float s2[256];
  const int tid = threadIdx.x;
  float a1 = 0.0f, a2 = 0.0f;
  for (int i = tid; i < NROWS; i += 256) {
    const float M = rowmax[i], S = rowsum[i], d = diag[i];
    a1 += d - (M + logf(S));
    float snd = S - expf(d - M);                    // drop diag term
    snd = fmaxf(snd, 1e-30f);
    a2 += d - (M + logf(snd));
  }
  s1[tid] = a1; s2[tid] = a2;
  __syncthreads();
  for (int st = 128; st > 0; st >>= 1) {
    if (tid < st) { s1[tid] += s1[tid + st]; s2[tid] += s2[tid + st]; }
    __syncthreads();
  }
  if (tid == 0) {
    const float B = (float)NROWS;
    out[0] = logf(B) + s1[0] / B;                   // infonce lower
    out[1] = s2[0] / B + logf(B - 1.0f);            // infonce upper (nodiag)
  }
}

// ---------------------------------------------------------------------------
extern "C" void kernel_launch(void* const* d_in, const int* in_sizes, int n_in,
                              void* d_out, int out_size, void* d_ws, size_t ws_size,
                              hipStream_t stream) {
  (void)in_sizes; (void)n_in; (void)out_size; (void)ws_size;
  const float* x   = (const float*)d_in[0];
  const float* y   = (const float*)d_in[1];
  const float* Wm1 = (const float*)d_in[2];  const float* bm1 = (const float*)d_in[3];
  const float* Wm2 = (const float*)d_in[4];  const float* bm2 = (const float*)d_in[5];
  const float* Wm3 = (const float*)d_in[6];  const float* bm3 = (const float*)d_in[7];
  const float* Wv1 = (const float*)d_in[8];  const float* bv1 = (const float*)d_in[9];
  const float* Wv2 = (const float*)d_in[10]; const float* bv2 = (const float*)d_in[11];
  const float* Wv3 = (const float*)d_in[12]; const float* bv3 = (const float*)d_in[13];

  // workspace (floats):
  //   Bc[8192*256] | cvec | rowmax | rowsum | diag | packed weights (327680)
  float* ws    = (float*)d_ws;
  float* Bc    = ws;
  float* cvec  = Bc + (size_t)NROWS * 256;
  float* rmx   = cvec + NROWS;
  float* rsm   = rmx + NROWS;
  float* dg    = rsm + NROWS;
  float* Wm1p  = dg + NROWS;
  float* Wm2p  = Wm1p + 256 * 256;
  float* Wm3p  = Wm2p + 256 * 256;
  float* Wv1p  = Wm3p + 256 * 128;
  float* Wv2p  = Wv1p + 256 * 256;
  float* Wv3p  = Wv2p + 256 * 256;   // + 256*128

  pack_w<<<256, 256, 0, stream>>>(Wm1, Wm1p, 256);
  pack_w<<<256, 256, 0, stream>>>(Wm2, Wm2p, 256);
  pack_w<<<128, 256, 0, stream>>>(Wm3, Wm3p, 128);
  pack_w<<<256, 256, 0, stream>>>(Wv1, Wv1p, 256);
  pack_w<<<256, 256, 0, stream>>>(Wv2, Wv2p, 256);
  pack_w<<<128, 256, 0, stream>>>(Wv3, Wv3p, 128);

  mlp_kernel<<<NROWS / 16, 256, 0, stream>>>(x, Wm1p, bm1, Wm2p, bm2, Wm3p, bm3,
                                             Wv1p, bv1, Wv2p, bv2, Wv3p, bv3,
                                             Bc, cvec);
  score_kernel<<<NROWS / 32, 256, 0, stream>>>(y, Bc, cvec, rmx, rsm, dg);
  reduce_kernel<<<1, 256, 0, stream>>>(rmx, rsm, dg, (float*)d_out);
}